// VFE_1391569404076
// MI455X (gfx1250) — compile-verified
//
#include <hip/hip_runtime.h>

typedef __attribute__((ext_vector_type(2))) float v2f;
typedef __attribute__((ext_vector_type(8))) float v8f;

constexpr int P  = 32;   // points per voxel
constexpr int IN = 7;    // input feature dim
constexpr int D2 = 64;   // half output dim
constexpr int VOX_PER_BLOCK = 4;   // 4 waves * 32 lanes = 128 threads

// One wave computes one voxel:
//   local[32x64] = x[32x7] @ W^T[7x64] + b   via V_WMMA_F32_16X16X4_F32
//   g[2q+e] = max over half-row q (reshape-view max semantics)
//   out row p (128 floats) = [ local[p, 0:64], g[0:64] ]
__global__ __launch_bounds__(128) void vfe_wmma_kernel(
    const float* __restrict__ x,     // [B, 32, 7]
    const float* __restrict__ W,     // [64, 7]
    const float* __restrict__ bias,  // [64]
    float* __restrict__ out)         // [B, 32, 128] flat
{
    __shared__ float lds_local[VOX_PER_BLOCK][P][D2];   // 4 * 8KB = 32KB
    __shared__ float lds_g[VOX_PER_BLOCK][D2];          // 1KB

    const int tid   = threadIdx.x;
    const int wave  = tid >> 5;      // 0..3
    const int lane  = tid & 31;
    const int lhalf = lane & 15;     // position within half-wave
    const int hi    = lane >> 4;     // 0 = lanes 0-15, 1 = lanes 16-31
    const int vox   = blockIdx.x * VOX_PER_BLOCK + wave;

    const float* xv = x + (size_t)vox * (P * IN);

    // ---- B fragments: B[k][n] = W[n*IN + k], K padded 7 -> 8 with zeros.
    // 32-bit B 4x16 layout (mirrors A 16x4): VGPR0 = K{0|2}, VGPR1 = K{1|3},
    // lanes 0-15 take K=0/1, lanes 16-31 take K=2/3 (+4 for second K-chunk).
    v2f bfrag0[4], bfrag1[4];
    float cbias[4];
    #pragma unroll
    for (int nt = 0; nt < 4; ++nt) {
        const int n = nt * 16 + lhalf;
        const float* wr = W + n * IN;
        const int k0 = hi * 2;
        v2f b0; b0.x = wr[k0];     b0.y = wr[k0 + 1];
        v2f b1; b1.x = wr[4 + k0]; b1.y = hi ? 0.0f : wr[5];  // k=7 pads to 0
        bfrag0[nt] = b0;
        bfrag1[nt] = b1;
        cbias[nt] = bias[n];   // per-column bias, broadcast across rows
    }

    // ---- GEMM: 2 M-tiles (points 0-15 / 16-31) x 4 N-tiles x 2 K-chunks
    #pragma unroll
    for (int mt = 0; mt < 2; ++mt) {
        const float* xr = xv + (mt * 16 + lhalf) * IN;   // A row for this lane
        const int k0 = hi * 2;
        v2f a0; a0.x = xr[k0];     a0.y = xr[k0 + 1];
        v2f a1; a1.x = xr[4 + k0]; a1.y = hi ? 0.0f : xr[5];  // k=7 pads to 0
        #pragma unroll
        for (int nt = 0; nt < 4; ++nt) {
            v8f acc;
            #pragma unroll
            for (int i = 0; i < 8; ++i) acc[i] = cbias[nt];
            acc = __builtin_amdgcn_wmma_f32_16x16x4_f32(
                false, a0, false, bfrag0[nt], (short)0, acc, false, false);
            acc = __builtin_amdgcn_wmma_f32_16x16x4_f32(
                false, a1, false, bfrag1[nt], (short)0, acc, false, false);
            // C/D layout: VGPR r -> row r (lanes 0-15) or r+8 (lanes 16-31)
            #pragma unroll
            for (int r = 0; r < 8; ++r) {
                lds_local[wave][mt * 16 + hi * 8 + r][nt * 16 + lhalf] = acc[r];
            }
        }
    }

    __syncthreads();

    // ---- reshape-view max: lane = point row; two 32-wide half-row maxes
    {
        const float* rowp = &lds_local[wave][lane][0];
        float mlo = rowp[0], mhi = rowp[32];
        #pragma unroll
        for (int c = 1; c < 32; ++c) {
            mlo = fmaxf(mlo, rowp[c]);
            mhi = fmaxf(mhi, rowp[32 + c]);
        }
        lds_g[wave][2 * lane]     = mlo;
        lds_g[wave][2 * lane + 1] = mhi;
    }

    __syncthreads();

    // ---- stream out: per point 128 floats = 32 lanes x float4 (coalesced)
    float* outb = out + (size_t)vox * (P * 2 * D2);
    float4 gvec;
    if (lane >= 16) gvec = *(const float4*)&lds_g[wave][(lane - 16) * 4];
    #pragma unroll
    for (int p = 0; p < P; ++p) {
        float4 v;
        if (lane < 16) v = *(const float4*)&lds_local[wave][p][lane * 4];
        else           v = gvec;
        ((float4*)(outb + p * 2 * D2))[lane] = v;
    }
}

extern "C" void kernel_launch(void* const* d_in, const int* in_sizes, int n_in,
                              void* d_out, int out_size, void* d_ws, size_t ws_size,
                              hipStream_t stream) {
    const float* x    = (const float*)d_in[0];
    const float* W    = (const float*)d_in[1];
    const float* bias = (const float*)d_in[2];
    float* out        = (float*)d_out;

    const int Bvox = in_sizes[0] / (P * IN);           // 32768
    const int grid = Bvox / VOX_PER_BLOCK;             // 8192 blocks
    vfe_wmma_kernel<<<grid, 128, 0, stream>>>(x, W, bias, out);
}